// S4DKernel_30794915512667
// MI455X (gfx1250) — compile-verified
//
#include <hip/hip_runtime.h>
#include <math.h>

typedef __attribute__((ext_vector_type(2))) float v2f;
typedef __attribute__((ext_vector_type(8))) float v8f;

#define H_DIM   1024
#define N2_DIM  32
#define L_LEN   4096
#define LHI     64          // outer l blocks (rows of A')
#define LLO     64          // inner l offset (cols of B')
#define KDIM    64          // 2 * N2 (re / -im interleaved)
#define APITCH  66          // 264 B/row: 8B-aligned rows, 2-bank stride -> conflict-free cols

// B pair-interleaved layout: Bq[slab = k0/4][half][l] = float2{B[kk][l], B[kk+1][l]}
// half-stride 160 dwords == 32 (mod 64 banks) -> the two wave halves cover
// complementary 32-bank sets -> conflict-free ds_load_b64 fragments.
#define BQ_HALF_STRIDE  160             // dwords (128 data + 32 pad)
#define BQ_SLAB_STRIDE  (2 * BQ_HALF_STRIDE)
#define BQ_TOTAL        (16 * BQ_SLAB_STRIDE)   // 5120 dwords = 20 KB

__device__ __forceinline__ int bq_idx(int slab, int half, int l) {
    return slab * BQ_SLAB_STRIDE + half * BQ_HALF_STRIDE + 2 * l;
}

// One block per head h. 128 threads = 4 wave32 waves.
// Per head: K[h, 64*lhi + llo] = 2 * Re( (Ck * V64^lhi) dot (V^llo) )
//         = 2 * (A' @ B')[lhi, llo]  with K-dim 64 (re / -im interleave).
__global__ __launch_bounds__(128) void s4d_wmma_kernel(
    const float* __restrict__ log_dt,
    const float* __restrict__ Cri,        // (H, N2, 2)
    const float* __restrict__ log_A_real, // (H, N2)
    const float* __restrict__ A_imag,     // (H, N2)
    float* __restrict__ out)              // (H, L)
{
    __shared__ float sdtA_r[N2_DIM];
    __shared__ float sdtA_i[N2_DIM];
    __shared__ float sCk_r[N2_DIM];
    __shared__ float sCk_i[N2_DIM];
    __shared__ float Alds[LHI][APITCH];   // A'[l_hi][2n+p]
    __shared__ float Bq[BQ_TOTAL];        // pair-interleaved B'

    const int h   = blockIdx.x;
    const int tid = threadIdx.x;

    // ---- Stage 1: per-mode coefficients dtA, Ck = Cc * (exp(dtA)-1)/A ----
    if (tid < N2_DIM) {
        const int n = tid;
        const float dt   = expf(log_dt[h]);
        const float A_re = -expf(log_A_real[h * N2_DIM + n]);
        const float A_im = A_imag[h * N2_DIM + n];
        const float dtA_r = dt * A_re;
        const float dtA_i = dt * A_im;
        // z = exp(dtA) - 1
        float s, c;
        const float e = expf(dtA_r);
        sincosf(dtA_i, &s, &c);
        const float z_r = e * c - 1.0f;
        const float z_i = e * s;
        // w = z / A  (complex divide)
        const float inv = 1.0f / (A_re * A_re + A_im * A_im);
        const float w_r = (z_r * A_re + z_i * A_im) * inv;
        const float w_i = (z_i * A_re - z_r * A_im) * inv;
        // Ck = Cc * w
        const float cr = Cri[(h * N2_DIM + n) * 2 + 0];
        const float ci = Cri[(h * N2_DIM + n) * 2 + 1];
        sCk_r[n]  = cr * w_r - ci * w_i;
        sCk_i[n]  = cr * w_i + ci * w_r;
        sdtA_r[n] = dtA_r;
        sdtA_i[n] = dtA_i;
    }
    __syncthreads();

    // ---- Stage 2: build A' (64 x 64, padded) and Bq in LDS ----
    // B row pair for mode n: {Re(exp(dtA*l)), -Im(exp(dtA*l))}
    //   rows (2n, 2n+1) -> pair kp = n -> slab n>>1, half n&1.
    for (int idx = tid; idx < N2_DIM * LLO; idx += 128) {
        const int n = idx >> 6;
        const int l = idx & 63;
        float s, c;
        const float er = expf(sdtA_r[n] * (float)l);
        sincosf(sdtA_i[n] * (float)l, &s, &c);
        v2f val;
        val.x =  er * c;
        val.y = -er * s;
        *(v2f*)&Bq[bq_idx(n >> 1, n & 1, l)] = val;
    }
    // A'[lhi][2n] = Re(Ck * exp(dtA*64*lhi)); A'[lhi][2n+1] = Im(...)
    for (int idx = tid; idx < N2_DIM * LHI; idx += 128) {
        const int lhi = idx & 63;
        const int n   = idx >> 6;
        const float t = (float)(64 * lhi);
        float s, c;
        const float er = expf(sdtA_r[n] * t);
        sincosf(sdtA_i[n] * t, &s, &c);
        const float wr = er * c;
        const float wi = er * s;
        Alds[lhi][2 * n + 0] = sCk_r[n] * wr - sCk_i[n] * wi;
        Alds[lhi][2 * n + 1] = sCk_r[n] * wi + sCk_i[n] * wr;
    }
    __syncthreads();

    // ---- Stage 3: 64x64x64 GEMM via V_WMMA_F32_16X16X4_F32 ----
    const int wave = tid >> 5;        // tile-row (l_hi block) owned by this wave
    const int lane = tid & 31;
    const int half = lane >> 4;       // 0: K pair {0,1}; 1: K pair {2,3}
    const int mrow = lane & 15;       // M (for A) / N (for B) within tile

    float* const o = out + (size_t)h * L_LEN;
    const int i = wave;               // tile row index 0..3
    const float* const arow = &Alds[16 * i + mrow][0];

    for (int j = 0; j < 4; ++j) {     // tile col index
        v8f acc = {};
        const int bcol = 16 * j + mrow;
#pragma unroll
        for (int k0 = 0; k0 < KDIM; k0 += 4) {
            // A frag: 16x4 f32 (ISA 7.12.2): lanes 0-15 K{0,1}, lanes 16-31 K{2,3}
            const v2f a = *(const v2f*)&arow[k0 + 2 * half];
            // B frag: single aligned 8B LDS load, no register shuffling
            const v2f b = *(const v2f*)&Bq[bq_idx(k0 >> 2, half, bcol)];
            acc = __builtin_amdgcn_wmma_f32_16x16x4_f32(
                /*neg_a=*/false, a, /*neg_b=*/false, b,
                /*c_mod=*/(short)0, acc, /*reuse_a=*/false, /*reuse_b=*/false);
        }
        // C layout: VGPR v -> M = v + 8*half, N = mrow (within tile)
        const int rb = 16 * i + 8 * half;
#pragma unroll
        for (int v = 0; v < 8; ++v) {
            o[(size_t)(rb + v) * LLO + bcol] = 2.0f * acc[v];
        }
    }
}

extern "C" void kernel_launch(void* const* d_in, const int* in_sizes, int n_in,
                              void* d_out, int out_size, void* d_ws, size_t ws_size,
                              hipStream_t stream) {
    (void)in_sizes; (void)n_in; (void)d_ws; (void)ws_size; (void)out_size;
    const float* log_dt     = (const float*)d_in[0];
    const float* C          = (const float*)d_in[1];
    const float* log_A_real = (const float*)d_in[2];
    const float* A_imag     = (const float*)d_in[3];
    // d_in[4] is L (== 4096), compile-time constant here.
    float* out = (float*)d_out;

    s4d_wmma_kernel<<<H_DIM, 128, 0, stream>>>(log_dt, C, log_A_real, A_imag, out);
}